// Brain_25288767439639
// MI455X (gfx1250) — compile-verified
//
#include <hip/hip_runtime.h>
#include <math.h>

// Problem constants fixed by the reference.
#define INPUT_SIZE_K  1024
#define OUTPUT_SIZE_K 256
#define STEPS_K       3

// ---------------------------------------------------------------------------
// Native f32 atomic add (no return) -> global_atomic_add_f32 on gfx1250.
// ---------------------------------------------------------------------------
__device__ __forceinline__ void atomic_add_f32(float* p, float v) {
#if defined(__HIP_DEVICE_COMPILE__)
    unsafeAtomicAdd(p, v);   // native FP32 atomic, not a CAS loop
#else
    atomicAdd(p, v);
#endif
}

__device__ __forceinline__ float fast_tanh(float a) {
#if __has_builtin(__builtin_amdgcn_tanhf)
    return __builtin_amdgcn_tanhf(a);   // CDNA5 v_tanh_f32 (TRANS pipe)
#else
    return tanhf(a);
#endif
}

// ---------------------------------------------------------------------------
// HOT KERNEL (placed first so the disasm snippet shows it): edge scatter.
// One quad of edges per wave32 lane:
//   3x global_load_b128 (src/dst/w streams, 12B/edge from HBM),
//   4 gathers of v   (400KB, L2-resident; scale_offset addressing),
//   4 no-return global_atomic_add_f32 into acc (400KB, RMW resolved in L2),
//   global_prefetch_b8 ~64KB ahead on each stream from 1 lane in 8.
// All indices kept 32-bit: E=10M so 4*t+3 < 2^31, and 32-bit VGPR indices
// let the compiler use gfx1250 scale_offset addressing everywhere.
// ---------------------------------------------------------------------------
__global__ void brain_scatter_kernel(const float* __restrict__ v,
                                     const float* __restrict__ w,
                                     const int*   __restrict__ src,
                                     const int*   __restrict__ dst,
                                     float* __restrict__ acc,
                                     int E) {
    const int t  = blockIdx.x * blockDim.x + threadIdx.x;
    const int i0 = t * 4;

    // Deepen the prefetch window for the three HBM streams. Speculative
    // prefetch past the end of the buffer is silently dropped (ISA 7.3).
    const int PF_ELEMS = 16384;  // 64KB ahead per stream
    if ((threadIdx.x & 7u) == 0u) {
        __builtin_prefetch(src + i0 + PF_ELEMS, 0, 1);
        __builtin_prefetch(dst + i0 + PF_ELEMS, 0, 1);
        __builtin_prefetch(w   + i0 + PF_ELEMS, 0, 1);
    }

    if (i0 + 3 < E) {
        const int4   s4 = *reinterpret_cast<const int4*>(src + i0);
        const int4   d4 = *reinterpret_cast<const int4*>(dst + i0);
        const float4 w4 = *reinterpret_cast<const float4*>(w + i0);

        atomic_add_f32(&acc[d4.x], v[s4.x] * w4.x);
        atomic_add_f32(&acc[d4.y], v[s4.y] * w4.y);
        atomic_add_f32(&acc[d4.z], v[s4.z] * w4.z);
        atomic_add_f32(&acc[d4.w], v[s4.w] * w4.w);
    } else {
        // Scalar tail (E not divisible by 4); no-op for fully OOB threads.
        for (int i = i0; i < E; ++i) {
            atomic_add_f32(&acc[dst[i]], v[src[i]] * w[i]);
        }
    }
}

// ---------------------------------------------------------------------------
// Initialize state v and the bias-preloaded accumulator.
//   v[i]   = x[i] for inputs, else 0
//   acc[i] = 0 for inputs, else bias[i - INPUT_SIZE]
// ---------------------------------------------------------------------------
__global__ void brain_init_kernel(const float* __restrict__ x,
                                  const float* __restrict__ bias,
                                  float* __restrict__ v,
                                  float* __restrict__ acc,
                                  int N) {
    int i = blockIdx.x * blockDim.x + threadIdx.x;
    if (i >= N) return;
    if (i < INPUT_SIZE_K) {
        v[i]   = x[i];
        acc[i] = 0.0f;
    } else {
        v[i]   = 0.0f;
        acc[i] = bias[i - INPUT_SIZE_K];
    }
}

// ---------------------------------------------------------------------------
// Activation + accumulator re-init (fused bias preload for the next step).
// On the final step, also emits the OUTPUT_SIZE tail to d_out.
// ---------------------------------------------------------------------------
__global__ void brain_act_kernel(float* __restrict__ v,
                                 float* __restrict__ acc,
                                 const float* __restrict__ bias,
                                 float* __restrict__ out,
                                 int N, int write_out) {
    int i = blockIdx.x * blockDim.x + threadIdx.x;
    if (i >= N) return;

    const float a = acc[i];
    float r;
    if (i < N - OUTPUT_SIZE_K) {
        r = fast_tanh(a);          // v_tanh_f32
    } else {
        r = a;                     // output neurons: no activation
        if (write_out) out[i - (N - OUTPUT_SIZE_K)] = r;
    }
    v[i] = r;

    // Reset accumulator (bias preload) for the next step.
    acc[i] = (i < INPUT_SIZE_K) ? 0.0f : bias[i - INPUT_SIZE_K];
}

// ---------------------------------------------------------------------------
// Host-side launcher.  Inputs (reference setup_inputs order):
//   d_in[0] x               [INPUT_SIZE]        float32
//   d_in[1] synapse_weights [E]                 float32
//   d_in[2] neuron_biases   [N - INPUT_SIZE]    float32
//   d_in[3] synapse_src     [E]                 int32
//   d_in[4] synapse_dst     [E]                 int32
// Output: d_out = v[N-OUTPUT_SIZE :]            float32 [OUTPUT_SIZE]
// Workspace: v (N floats) + acc (N floats) = 800KB of d_ws.
// ---------------------------------------------------------------------------
extern "C" void kernel_launch(void* const* d_in, const int* in_sizes, int n_in,
                              void* d_out, int out_size, void* d_ws, size_t ws_size,
                              hipStream_t stream) {
    const float* x    = (const float*)d_in[0];
    const float* w    = (const float*)d_in[1];
    const float* bias = (const float*)d_in[2];
    const int*   src  = (const int*)d_in[3];
    const int*   dst  = (const int*)d_in[4];
    float*       out  = (float*)d_out;

    const int E = in_sizes[1];
    const int N = in_sizes[2] + INPUT_SIZE_K;   // biases cover N - INPUT_SIZE

    float* v   = (float*)d_ws;
    float* acc = v + N;

    const int threads = 256;                    // 8 wave32 waves per block
    const int gridN   = (N + threads - 1) / threads;
    const long long quads = ((long long)E + 3) / 4;
    const int gridE   = (int)((quads + (long long)threads - 1) / threads);

    brain_init_kernel<<<gridN, threads, 0, stream>>>(x, bias, v, acc, N);

    for (int step = 0; step < STEPS_K; ++step) {
        brain_scatter_kernel<<<gridE, threads, 0, stream>>>(v, w, src, dst, acc, E);
        brain_act_kernel<<<gridN, threads, 0, stream>>>(
            v, acc, bias, out, N, (step == STEPS_K - 1) ? 1 : 0);
    }
}